// SVDHead_27754078667078
// MI455X (gfx1250) — compile-verified
//
#include <hip/hip_runtime.h>
#include <hip/hip_bf16.h>

typedef __attribute__((ext_vector_type(2))) float v2f;
typedef __attribute__((ext_vector_type(8))) float v8f;

#define CHUNKS 64
#define TPB    256

// ---------------------------------------------------------------------------
// Stage 1: memory-bound streaming reduction.
// Grid: (CHUNKS, B) blocks x 256 threads. Each thread handles 4 consecutive
// points via float4 (global_load_b128), accumulates 16 products:
//   rows {w*sx, w*sy, w*sz, w} x cols {tx, ty, tz, 1}
// then LDS tree-reduce to one 16-vector per block -> workspace.
// ---------------------------------------------------------------------------
__global__ void kabsch_partial(const float* __restrict__ src,
                               const float* __restrict__ tgt,
                               const float* __restrict__ wgt,
                               float* __restrict__ part, int N) {
  const int chunk = blockIdx.x;
  const int b     = blockIdx.y;
  const int tid   = threadIdx.x;
  const int npts  = N / CHUNKS;                       // 1024 points per block
  const size_t base = (size_t)chunk * npts + (size_t)tid * 4;

  const size_t sb = (size_t)b * 3 * N;
  const float4 sx = *(const float4*)(src + sb + 0 * (size_t)N + base);
  const float4 sy = *(const float4*)(src + sb + 1 * (size_t)N + base);
  const float4 sz = *(const float4*)(src + sb + 2 * (size_t)N + base);
  const float4 tx = *(const float4*)(tgt + sb + 0 * (size_t)N + base);
  const float4 ty = *(const float4*)(tgt + sb + 1 * (size_t)N + base);
  const float4 tz = *(const float4*)(tgt + sb + 2 * (size_t)N + base);
  const float4 wv = *(const float4*)(wgt + (size_t)b * N + base);

  float acc[16];
#pragma unroll
  for (int f = 0; f < 16; ++f) acc[f] = 0.0f;

#define ACCUM(SX, SY, SZ, TX, TY, TZ, WW) do {                         \
    float r0 = (WW) * (SX), r1 = (WW) * (SY), r2 = (WW) * (SZ), r3 = (WW); \
    acc[0]  += r0 * (TX); acc[1]  += r0 * (TY); acc[2]  += r0 * (TZ); acc[3]  += r0; \
    acc[4]  += r1 * (TX); acc[5]  += r1 * (TY); acc[6]  += r1 * (TZ); acc[7]  += r1; \
    acc[8]  += r2 * (TX); acc[9]  += r2 * (TY); acc[10] += r2 * (TZ); acc[11] += r2; \
    acc[12] += r3 * (TX); acc[13] += r3 * (TY); acc[14] += r3 * (TZ); acc[15] += r3; \
  } while (0)

  ACCUM(sx.x, sy.x, sz.x, tx.x, ty.x, tz.x, wv.x);
  ACCUM(sx.y, sy.y, sz.y, tx.y, ty.y, tz.y, wv.y);
  ACCUM(sx.z, sy.z, sz.z, tx.z, ty.z, tz.z, wv.z);
  ACCUM(sx.w, sy.w, sz.w, tx.w, ty.w, tz.w, wv.w);
#undef ACCUM

  __shared__ float red[16][TPB];
#pragma unroll
  for (int f = 0; f < 16; ++f) red[f][tid] = acc[f];
  __syncthreads();

  for (int s = TPB / 2; s > 0; s >>= 1) {
    if (tid < s) {
#pragma unroll
      for (int f = 0; f < 16; ++f) red[f][tid] += red[f][tid + s];
    }
    __syncthreads();
  }

  if (tid < 16)
    part[((size_t)b * CHUNKS + chunk) * 16 + tid] = red[tid][0];
}

// ---------------------------------------------------------------------------
// Stage 2: one wave32 per batch.
// WMMA reduction: D(16x16) += A(16x4) * B(4x16) with A = ones in row 0 only,
// B rows = 4 chunk partial 16-vectors per issue.  After 16 issues D[0][f] is
// the full-batch sum of feature f (f32 end to end).  Then lane 0 runs the
// 3x3 Kabsch closed form (Jacobi eigensolve of H^T H).
// ---------------------------------------------------------------------------
__global__ void kabsch_finalize(const float* __restrict__ part,
                                float* __restrict__ out, int B) {
  const int b    = blockIdx.x;
  const int lane = threadIdx.x;                 // 0..31, full wave active

  // A-matrix 16x4: lane l<16 holds A[l][0..1], lane l>=16 holds A[l-16][2..3].
  // Row 0 = ones, rows 1..15 = 0  ->  only lanes 0 and 16 carry ones.
  const float aval = ((lane & 15) == 0) ? 1.0f : 0.0f;
  v2f amat = {aval, aval};

  // B-matrix 4x16: lane l<16 holds B[0..1][l], lane l>=16 holds B[2..3][l-16].
  const int col   = lane & 15;
  const int khalf = (lane >> 4) * 2;            // 0 or 2
  const float* Pb = part + (size_t)b * CHUNKS * 16;

  v8f acc = {0.0f, 0.0f, 0.0f, 0.0f, 0.0f, 0.0f, 0.0f, 0.0f};
#pragma unroll
  for (int c = 0; c < CHUNKS; c += 4) {
    v2f bmat;
    bmat.x = Pb[(size_t)(c + khalf + 0) * 16 + col];
    bmat.y = Pb[(size_t)(c + khalf + 1) * 16 + col];
    acc = __builtin_amdgcn_wmma_f32_16x16x4_f32(
        /*neg_a=*/false, amat, /*neg_b=*/false, bmat,
        /*c_mod=*/(short)0, acc, /*reuse_a=*/false, /*reuse_b=*/false);
  }

  // C/D layout: VGPR0, lanes 0-15 = row M=0, N=lane.
  __shared__ float sums[16];
  if (lane < 16) sums[lane] = acc[0];
  __syncthreads();

  if (lane == 0) {
    float M[3][3], sm[3], tm[3];
#pragma unroll
    for (int i = 0; i < 3; ++i) {
#pragma unroll
      for (int j = 0; j < 3; ++j) M[i][j] = sums[i * 4 + j];
      sm[i] = sums[i * 4 + 3];
      tm[i] = sums[12 + i];
    }
    const float W = sums[15];

    // H = sum w (s - sm)(t - tm)^T = M + (W - 2) * sm tm^T
    float H[3][3];
#pragma unroll
    for (int i = 0; i < 3; ++i)
#pragma unroll
      for (int j = 0; j < 3; ++j)
        H[i][j] = M[i][j] + (W - 2.0f) * sm[i] * tm[j];

    // A = H^T H (symmetric), Jacobi eigensolve -> V (right singular vectors)
    float A[3][3];
#pragma unroll
    for (int i = 0; i < 3; ++i)
#pragma unroll
      for (int j = 0; j < 3; ++j)
        A[i][j] = H[0][i] * H[0][j] + H[1][i] * H[1][j] + H[2][i] * H[2][j];

    float V[3][3] = {{1.f, 0.f, 0.f}, {0.f, 1.f, 0.f}, {0.f, 0.f, 1.f}};
    const int ps[3] = {0, 0, 1}, qs[3] = {1, 2, 2};
    for (int sweep = 0; sweep < 10; ++sweep) {
      for (int r = 0; r < 3; ++r) {
        const int p = ps[r], q = qs[r];
        const float apq = A[p][q];
        if (fabsf(apq) > 1e-30f) {
          const float tau = (A[q][q] - A[p][p]) / (2.0f * apq);
          const float tj  = (tau >= 0.0f ? 1.0f : -1.0f) /
                            (fabsf(tau) + sqrtf(1.0f + tau * tau));
          const float cth = 1.0f / sqrtf(1.0f + tj * tj);
          const float sth = tj * cth;
#pragma unroll
          for (int k = 0; k < 3; ++k) {
            const float akp = A[k][p], akq = A[k][q];
            A[k][p] = cth * akp - sth * akq;
            A[k][q] = sth * akp + cth * akq;
          }
#pragma unroll
          for (int k = 0; k < 3; ++k) {
            const float apk = A[p][k], aqk = A[q][k];
            A[p][k] = cth * apk - sth * aqk;
            A[q][k] = sth * apk + cth * aqk;
          }
#pragma unroll
          for (int k = 0; k < 3; ++k) {
            const float vkp = V[k][p], vkq = V[k][q];
            V[k][p] = cth * vkp - sth * vkq;
            V[k][q] = sth * vkp + cth * vkq;
          }
        }
      }
    }

    float lam[3] = {A[0][0], A[1][1], A[2][2]};
    // sort eigenvalues descending, permuting V columns
    for (int i = 0; i < 2; ++i)
      for (int j = 0; j < 2 - i; ++j)
        if (lam[j] < lam[j + 1]) {
          const float tl = lam[j]; lam[j] = lam[j + 1]; lam[j + 1] = tl;
          for (int k = 0; k < 3; ++k) {
            const float tv = V[k][j]; V[k][j] = V[k][j + 1]; V[k][j + 1] = tv;
          }
        }

    const float s0 = fmaxf(sqrtf(fmaxf(lam[0], 0.0f)), 1e-20f);
    const float s1 = fmaxf(sqrtf(fmaxf(lam[1], 0.0f)), 1e-20f);

    float u0[3], u1[3], u2[3];
#pragma unroll
    for (int i = 0; i < 3; ++i) {
      u0[i] = (H[i][0] * V[0][0] + H[i][1] * V[1][0] + H[i][2] * V[2][0]) / s0;
      u1[i] = (H[i][0] * V[0][1] + H[i][1] * V[1][1] + H[i][2] * V[2][1]) / s1;
    }
    // light re-orthonormalization of U's first two columns
    float n0 = fmaxf(sqrtf(u0[0]*u0[0] + u0[1]*u0[1] + u0[2]*u0[2]), 1e-20f);
#pragma unroll
    for (int i = 0; i < 3; ++i) u0[i] /= n0;
    const float d01 = u0[0]*u1[0] + u0[1]*u1[1] + u0[2]*u1[2];
#pragma unroll
    for (int i = 0; i < 3; ++i) u1[i] -= d01 * u0[i];
    float n1 = fmaxf(sqrtf(u1[0]*u1[0] + u1[1]*u1[1] + u1[2]*u1[2]), 1e-20f);
#pragma unroll
    for (int i = 0; i < 3; ++i) u1[i] /= n1;
    // u2 = u0 x u1  ->  det(U) = +1
    u2[0] = u0[1]*u1[2] - u0[2]*u1[1];
    u2[1] = u0[2]*u1[0] - u0[0]*u1[2];
    u2[2] = u0[0]*u1[1] - u0[1]*u1[0];

    // force det(V) = +1 (flip column of smallest singular value, as reference)
    const float detV =
        V[0][0] * (V[1][1] * V[2][2] - V[1][2] * V[2][1]) -
        V[0][1] * (V[1][0] * V[2][2] - V[1][2] * V[2][0]) +
        V[0][2] * (V[1][0] * V[2][1] - V[1][1] * V[2][0]);
    if (detV < 0.0f) {
#pragma unroll
      for (int k = 0; k < 3; ++k) V[k][2] = -V[k][2];
    }

    // R = V U^T ; t = tgtm - R srcm
    float R[3][3];
#pragma unroll
    for (int i = 0; i < 3; ++i)
#pragma unroll
      for (int j = 0; j < 3; ++j)
        R[i][j] = V[i][0] * u0[j] + V[i][1] * u1[j] + V[i][2] * u2[j];

#pragma unroll
    for (int i = 0; i < 3; ++i) {
#pragma unroll
      for (int j = 0; j < 3; ++j)
        out[(size_t)b * 9 + i * 3 + j] = R[i][j];
      out[(size_t)B * 9 + (size_t)b * 3 + i] =
          tm[i] - (R[i][0] * sm[0] + R[i][1] * sm[1] + R[i][2] * sm[2]);
    }
  }
}

extern "C" void kernel_launch(void* const* d_in, const int* in_sizes, int n_in,
                              void* d_out, int out_size, void* d_ws, size_t ws_size,
                              hipStream_t stream) {
  const float* src = (const float*)d_in[0];   // [B,3,N] f32
  const float* tgt = (const float*)d_in[1];   // [B,3,N] f32
  const float* wgt = (const float*)d_in[2];   // [B,N]   f32

  const int N = 65536;
  const int B = in_sizes[2] / N;              // 64

  float* part = (float*)d_ws;                 // B*CHUNKS*16 floats = 256 KB

  dim3 g1(CHUNKS, B);
  kabsch_partial<<<g1, TPB, 0, stream>>>(src, tgt, wgt, part, N);
  kabsch_finalize<<<B, 32, 0, stream>>>(part, (float*)d_out, B);
}